// BaselineParameterizedPool2D_32727650796057
// MI455X (gfx1250) — compile-verified
//
#include <hip/hip_runtime.h>

typedef unsigned int v4u __attribute__((ext_vector_type(4)));
typedef int          v8i __attribute__((ext_vector_type(8)));
typedef int          v4i __attribute__((ext_vector_type(4)));

#define BATCH   16
#define CHANS   64
#define HIN     256
#define WIN     256
#define HOUT    128
#define WOUT    128
#define PADV    (-10.0f)
#define ROWS    16                 // output rows per workgroup tile
#define IN_ROWS (2 * ROWS + 1)     // 33 input rows staged in LDS
#define ROWS_PER_WAVE 5            // 8 waves * 5 >= 33

__global__ __launch_bounds__(256)
void pbias_maxpool_tdm(const float* __restrict__ xs,
                       const float* __restrict__ h,
                       float* __restrict__ pooled,
                       int*   __restrict__ prov) {
    __shared__ float tile[IN_ROWS * WIN];   // 33 * 256 * 4 = 33 KB

    const int tid      = threadIdx.x;
    const int tileI    = blockIdx.x;            // 0..7 row tile
    const int plane    = blockIdx.y;            // b*C + c
    const int c        = plane & (CHANS - 1);
    const int ho0      = tileI * ROWS;
    const int row_base = 2 * ho0 - 1;           // input row of LDS slot 0

    // ---------------- TDM async stage: each wave DMA's its 5-row slice ----
    const int wave = __builtin_amdgcn_readfirstlane(tid >> 5);
    int s0 = wave * ROWS_PER_WAVE;
    int s1 = s0 + ROWS_PER_WAVE; if (s1 > IN_ROWS) s1 = IN_ROWS;
    int lo = -row_base; if (lo < 0) lo = 0;
    if (s0 < lo) s0 = lo;
    int hiex = HIN - row_base;                  // exclusive valid slot bound
    if (s1 > hiex) s1 = hiex;
    int n = s1 - s0; if (n < 0) n = 0;

    unsigned lds_addr = (unsigned)(size_t)(&tile[s0 * WIN]);
    unsigned long long gaddr =
        (unsigned long long)(size_t)(xs + ((size_t)plane * (HIN * (size_t)WIN)
                                           + (size_t)(row_base + s0) * WIN));

    // D# group 0: count=1 | lds_addr | global_addr | type=2
    v4u g0;
    g0.x = 1u;
    g0.y = lds_addr;
    g0.z = (unsigned)gaddr;
    g0.w = ((unsigned)(gaddr >> 32) & 0x01FFFFFFu) | (2u << 30);

    // D# group 1: 2-D tile, 4-byte elements, row stride = WIN
    const int tile0 = (n > 0) ? WIN : 0;        // tile_dim0 = 0 -> NOP
    v8i g1;
    g1[0] = (2 << 16);                          // data_size = 4 bytes
    g1[1] = (WIN   & 0xFFFF) << 16;             // tensor_dim0[15:0]
    g1[2] = (n     & 0xFFFF) << 16;             // tensor_dim1[15:0]
    g1[3] = (tile0 & 0xFFFF) << 16;             // tile_dim0
    g1[4] = (n     & 0xFFFF);                   // tile_dim1 (tile_dim2 = 0)
    g1[5] = WIN;                                // tensor_dim0_stride low
    g1[6] = 0;
    g1[7] = 0;
    v4i z4 = {0, 0, 0, 0};
#if __has_include(<hip/amd_detail/amd_gfx1250_TDM.h>)
    v8i z8 = {0, 0, 0, 0, 0, 0, 0, 0};
    __builtin_amdgcn_tensor_load_to_lds(g0, g1, z4, z4, z8, 0);
#else
    __builtin_amdgcn_tensor_load_to_lds(g0, g1, z4, z4, 0);
#endif

    // Per-channel tap biases (tap k = di*3 + dj, matching torch Unfold order)
    float bias[9];
#pragma unroll
    for (int t = 0; t < 9; ++t) bias[t] = h[t * CHANS + c];

    __builtin_amdgcn_s_wait_tensorcnt(0);
    __syncthreads();

    // ---------------- compute: 8 output pixels per thread ------------------
    const int wo   = tid & (WOUT - 1);
    const int rsel = tid >> 7;                  // 0 or 1
    const int colm1 = (wo == 0) ? 0 : (2 * wo - 1);  // clamped left-tap column
    const bool leftok = (wo > 0);               // wi = -1 only at wo == 0

#pragma unroll
    for (int k = 0; k < ROWS / 2; ++k) {
        const int hl = rsel + 2 * k;            // local output row 0..15
        float best = -3.4028235e38f;
        int   arg  = 0;
#pragma unroll
        for (int di = 0; di < 3; ++di) {
            const int s  = 2 * hl + di;         // LDS slot (0..32)
            const int gy = row_base + s;        // global input row
            const bool rowok = (gy >= 0);       // gy <= 255 always by tiling

            // columns 2wo and 2wo+1 are always in-bounds and 8B-aligned
            const float2 p = *(const float2*)&tile[s * WIN + 2 * wo];
            const float  l = tile[s * WIN + colm1];

            const float t0 = (rowok && leftok) ? l   : PADV;
            const float t1 = rowok             ? p.x : PADV;
            const float t2 = rowok             ? p.y : PADV;

            const float v0 = t0 + bias[di * 3 + 0];
            const float v1 = t1 + bias[di * 3 + 1];
            const float v2 = t2 + bias[di * 3 + 2];

            if (v0 > best) { best = v0; arg = di * 3 + 0; }  // first-wins ties
            if (v1 > best) { best = v1; arg = di * 3 + 1; }
            if (v2 > best) { best = v2; arg = di * 3 + 2; }
        }
        const size_t o = ((size_t)plane * HOUT + (size_t)(ho0 + hl)) * WOUT + wo;
        __builtin_nontemporal_store(best, &pooled[o]);
        __builtin_nontemporal_store(arg, &prov[o]);
    }
}

extern "C" void kernel_launch(void* const* d_in, const int* in_sizes, int n_in,
                              void* d_out, int out_size, void* d_ws, size_t ws_size,
                              hipStream_t stream) {
    (void)in_sizes; (void)n_in; (void)out_size; (void)d_ws; (void)ws_size;
    const float* xs = (const float*)d_in[0];
    const float* h  = (const float*)d_in[1];
    float* pooled = (float*)d_out;
    int*   prov   = (int*)d_out + (size_t)BATCH * CHANS * HOUT * WOUT;

    dim3 grid(HOUT / ROWS, BATCH * CHANS);     // (8, 1024)
    pbias_maxpool_tdm<<<grid, 256, 0, stream>>>(xs, h, pooled, prov);
}